// GNNFeatureExtractor_58342835749498
// MI455X (gfx1250) — compile-verified
//
#include <hip/hip_runtime.h>

#define HIDDIM 128
#define NTILE  8     // 8 x 16-wide output column tiles = 128 cols
#define WAVES  4     // one wave per SIMD32 of the WGP

typedef float v2f __attribute__((ext_vector_type(2)));
typedef float v8f __attribute__((ext_vector_type(8)));

__device__ __forceinline__ v8f wmma_f32_k4(v2f a, v2f b, v8f c) {
  // D = A(16x4,f32) x B(4x16,f32) + C(16x16,f32), exact fp32 matrix pipe
  return __builtin_amdgcn_wmma_f32_16x16x4_f32(false, a, false, b, (short)0, c,
                                               false, false);
}

// CDNA5 async DMA: global memory -> LDS, 16B per lane, tracked by ASYNCcnt.
// VDST operand = per-lane LDS byte address (low 32 bits of generic pointer),
// VADDR = 64-bit global address (GV mode, no SADDR).
__device__ __forceinline__ void async_copy_b128(const float* gsrc,
                                                float* lds_dst) {
  unsigned lofs = (unsigned)(unsigned long long)lds_dst;
  asm volatile("global_load_async_to_lds_b128 %0, %1, off"
               :
               : "v"(lofs), "v"(gsrc)
               : "memory");
}
__device__ __forceinline__ void async_wait0() {
  asm volatile("s_wait_asynccnt 0x0" ::: "memory");
}

// ---------------------------------------------------------------- utilities
__global__ void zero_f32(float* __restrict__ p, long long n) {
  long long i = (long long)blockIdx.x * blockDim.x + threadIdx.x;
  if (i < n) p[i] = 0.f;
}

__global__ void deg_acc_kernel(const long long* __restrict__ edge_index,
                               float* __restrict__ deg, int E) {
  int e = blockIdx.x * blockDim.x + threadIdx.x;
  if (e < E) atomicAdd(&deg[edge_index[(long long)E + e]], 1.f);
}

__global__ void deg_inv_kernel(float* __restrict__ deg, int n) {
  int i = blockIdx.x * blockDim.x + threadIdx.x;
  if (i < n) deg[i] = 1.f / fmaxf(deg[i], 1.f);
}

// x = node_feat @ enc_W + enc_b   ([N,5] @ [5,128]); K=5 -> scalar FMA kernel
__global__ void encoder_kernel(const float* __restrict__ nf,
                               const float* __restrict__ W,
                               const float* __restrict__ b,
                               float* __restrict__ x, int N) {
  long long t = (long long)blockIdx.x * blockDim.x + threadIdx.x;
  if (t >= (long long)N * HIDDIM) return;
  int i = (int)(t >> 7), h = (int)(t & 127);
  const float* r = nf + (long long)i * 5;
  float acc = b[h];
#pragma unroll
  for (int k = 0; k < 5; ++k) acc += r[k] * W[k * HIDDIM + h];
  x[t] = acc;
}

// ------------------------------------------------- edge message MLP + scatter
// msg = relu([x_dst|x_src|ea] @ W1 + b1) @ W2 + b2 ; aggr[dst] += msg*inv_deg
__global__ __launch_bounds__(WAVES * 32, 1)
void edge_msg_kernel(const float* __restrict__ x,
                     const float* __restrict__ edge_attr,
                     const long long* __restrict__ edge_index,
                     const float* __restrict__ W1g, const float* __restrict__ b1g,
                     const float* __restrict__ W2g, const float* __restrict__ b2g,
                     const float* __restrict__ inv_deg,
                     float* __restrict__ aggr, int n_edges) {
  extern __shared__ float smem[];
  float* W1T     = smem;                        // [128][260] transposed, row 256 = ea weight
  float* W2T     = W1T + 128 * 260;             // [128][132] transposed
  float* AsubAll = W2T + 128 * 132;             // WAVES x [16][256] staged m_in
  float* HsubAll = AsubAll + WAVES * 16 * 256;  // WAVES x [16][132] hidden act
  float* metaAll = HsubAll + WAVES * 16 * 132;  // WAVES x 32: [0..15]=scale [16..31]=ea
  int*   dstAll  = (int*)(metaAll + WAVES * 32);

  const int tid = threadIdx.x;
  // cooperative transposed weight staging (coalesced global reads per k-row)
  for (int k = 0; k < 257; ++k) W1T[tid * 260 + k] = W1g[k * HIDDIM + tid];
  for (int k = 0; k < 128; ++k) W2T[tid * 132 + k] = W2g[k * HIDDIM + tid];
  __syncthreads();

  const int wave   = tid >> 5;
  const int lane   = tid & 31;
  const int n      = lane & 15;          // B col (and A row m) for this lane
  const int kg     = (lane >> 4) << 1;   // K sub-offset: 0 or 2
  const int rowoff = (lane >> 4) << 3;   // C/D row group: 0 or 8

  float* Asub = AsubAll + wave * 16 * 256;
  float* Hsub = HsubAll + wave * 16 * 132;
  float* meta = metaAll + wave * 32;
  int*   dstw = dstAll + wave * 16;

  float b1v[NTILE], b2v[NTILE], wEv[NTILE];
#pragma unroll
  for (int nt = 0; nt < NTILE; ++nt) {
    b1v[nt] = b1g[nt * 16 + n];
    b2v[nt] = b2g[nt * 16 + n];
    wEv[nt] = W1T[(nt * 16 + n) * 260 + 256];  // edge_attr weight row
  }

  const int ntiles = n_edges >> 4;
  for (int tile = blockIdx.x * WAVES + wave; tile < ntiles;
       tile += gridDim.x * WAVES) {
    const int e0 = tile << 4;
    // async DMA gather of 16 edges' [x_dst | x_src] rows (L2-resident):
    // memory -> LDS directly, no VGPR round-trip; tracked by ASYNCcnt
    for (int r = 0; r < 16; ++r) {
      const long long s = edge_index[e0 + r];
      const long long d = edge_index[(long long)n_edges + e0 + r];
      async_copy_b128(x + d * HIDDIM + lane * 4, Asub + r * 256 + lane * 4);
      async_copy_b128(x + s * HIDDIM + lane * 4,
                      Asub + r * 256 + 128 + lane * 4);
    }
    if (lane < 16) {
      const long long d = edge_index[(long long)n_edges + e0 + lane];
      dstw[lane]      = (int)d;
      meta[lane]      = inv_deg[d];
      meta[16 + lane] = edge_attr[e0 + lane];
    }
    async_wait0();  // A tile resident in LDS

    // ---- GEMM1: hidden = relu(m_in @ W1 + b1); ea column folded into C init
    v8f c[NTILE];
#pragma unroll
    for (int nt = 0; nt < NTILE; ++nt) {
#pragma unroll
      for (int j = 0; j < 8; ++j)
        c[nt][j] = b1v[nt] + meta[16 + rowoff + j] * wEv[nt];
    }
    for (int kk = 0; kk < 64; ++kk) {
      const int k0 = (kk << 2) + kg;
      const v2f a = *(const v2f*)(Asub + n * 256 + k0);
#pragma unroll
      for (int nt = 0; nt < NTILE; ++nt) {
        const v2f b = *(const v2f*)(W1T + (nt * 16 + n) * 260 + k0);
        c[nt] = wmma_f32_k4(a, b, c[nt]);
      }
    }
#pragma unroll
    for (int nt = 0; nt < NTILE; ++nt)
#pragma unroll
      for (int j = 0; j < 8; ++j)
        Hsub[(rowoff + j) * 132 + nt * 16 + n] = fmaxf(c[nt][j], 0.f);

    // ---- GEMM2: msg = hidden @ W2 + b2
#pragma unroll
    for (int nt = 0; nt < NTILE; ++nt)
#pragma unroll
      for (int j = 0; j < 8; ++j) c[nt][j] = b2v[nt];
    for (int kk = 0; kk < 32; ++kk) {
      const int k0 = (kk << 2) + kg;
      const v2f a = *(const v2f*)(Hsub + n * 132 + k0);
#pragma unroll
      for (int nt = 0; nt < NTILE; ++nt) {
        const v2f b = *(const v2f*)(W2T + (nt * 16 + n) * 132 + k0);
        c[nt] = wmma_f32_k4(a, b, c[nt]);
      }
    }
    // scatter-mean: inv_deg folded before the sum (same result as mean-after)
#pragma unroll
    for (int j = 0; j < 8; ++j) {
      const int mr = rowoff + j;
      const float s = meta[mr];
      float* dstp = aggr + (size_t)dstw[mr] * HIDDIM;
#pragma unroll
      for (int nt = 0; nt < NTILE; ++nt)
        atomicAdd(dstp + nt * 16 + n, c[nt][j] * s);
    }
  }
}

// ------------------------------------------- node update MLP + LayerNorm+ReLU
__global__ __launch_bounds__(WAVES * 32, 1)
void node_update_kernel(const float* __restrict__ x,
                        const float* __restrict__ aggr,
                        const float* __restrict__ W1g, const float* __restrict__ b1g,
                        const float* __restrict__ W2g, const float* __restrict__ b2g,
                        const float* __restrict__ lng, const float* __restrict__ lnb,
                        float* __restrict__ xout, int n_nodes) {
  extern __shared__ float smem[];
  float* W1T     = smem;                        // [128][260] (256 cols used)
  float* W2T     = W1T + 128 * 260;             // [128][132]
  float* AsubAll = W2T + 128 * 132;             // WAVES x [16][256]
  float* HsubAll = AsubAll + WAVES * 16 * 256;  // WAVES x [16][132]

  const int tid = threadIdx.x;
  for (int k = 0; k < 256; ++k) W1T[tid * 260 + k] = W1g[k * HIDDIM + tid];
  for (int k = 0; k < 128; ++k) W2T[tid * 132 + k] = W2g[k * HIDDIM + tid];
  __syncthreads();

  const int wave = tid >> 5, lane = tid & 31;
  const int n = lane & 15;
  const int kg = (lane >> 4) << 1;
  const int rowoff = (lane >> 4) << 3;
  float* Asub = AsubAll + wave * 16 * 256;
  float* Hsub = HsubAll + wave * 16 * 132;

  float b1v[NTILE], b2v[NTILE];
#pragma unroll
  for (int nt = 0; nt < NTILE; ++nt) {
    b1v[nt] = b1g[nt * 16 + n];
    b2v[nt] = b2g[nt * 16 + n];
  }

  const int ntiles = n_nodes >> 4;
  for (int tile = blockIdx.x * WAVES + wave; tile < ntiles;
       tile += gridDim.x * WAVES) {
    const int node0 = tile << 4;
    // async DMA staging of [x | aggr] rows into LDS
    for (int r = 0; r < 16; ++r) {
      async_copy_b128(x + (size_t)(node0 + r) * HIDDIM + lane * 4,
                      Asub + r * 256 + lane * 4);
      async_copy_b128(aggr + (size_t)(node0 + r) * HIDDIM + lane * 4,
                      Asub + r * 256 + 128 + lane * 4);
    }
    async_wait0();

    // GEMM1: h = relu([x|aggr] @ W1u + b1u), K=256
    v8f c[NTILE];
#pragma unroll
    for (int nt = 0; nt < NTILE; ++nt)
#pragma unroll
      for (int j = 0; j < 8; ++j) c[nt][j] = b1v[nt];
    for (int kk = 0; kk < 64; ++kk) {
      const int k0 = (kk << 2) + kg;
      const v2f a = *(const v2f*)(Asub + n * 256 + k0);
#pragma unroll
      for (int nt = 0; nt < NTILE; ++nt) {
        const v2f b = *(const v2f*)(W1T + (nt * 16 + n) * 260 + k0);
        c[nt] = wmma_f32_k4(a, b, c[nt]);
      }
    }
#pragma unroll
    for (int nt = 0; nt < NTILE; ++nt)
#pragma unroll
      for (int j = 0; j < 8; ++j)
        Hsub[(rowoff + j) * 132 + nt * 16 + n] = fmaxf(c[nt][j], 0.f);

    // GEMM2: x_new = h @ W2u + b2u, K=128 -> stash rows in Asub (free now)
#pragma unroll
    for (int nt = 0; nt < NTILE; ++nt)
#pragma unroll
      for (int j = 0; j < 8; ++j) c[nt][j] = b2v[nt];
    for (int kk = 0; kk < 32; ++kk) {
      const int k0 = (kk << 2) + kg;
      const v2f a = *(const v2f*)(Hsub + n * 132 + k0);
#pragma unroll
      for (int nt = 0; nt < NTILE; ++nt) {
        const v2f b = *(const v2f*)(W2T + (nt * 16 + n) * 132 + k0);
        c[nt] = wmma_f32_k4(a, b, c[nt]);
      }
    }
#pragma unroll
    for (int nt = 0; nt < NTILE; ++nt)
#pragma unroll
      for (int j = 0; j < 8; ++j)
        Asub[(rowoff + j) * 132 + nt * 16 + n] = c[nt][j];

    // LayerNorm + ReLU: row r handled by lane pair {r, r+16} via shfl_xor(16)
    const int r = lane & 15;
    const int ch = (lane >> 4) << 6;  // cols 0..63 or 64..127
    float s = 0.f, ss = 0.f;
    for (int kc = 0; kc < 64; ++kc) {
      const float v = Asub[r * 132 + ch + kc];
      s += v;
      ss += v * v;
    }
    s += __shfl_xor(s, 16);
    ss += __shfl_xor(ss, 16);
    const float mu = s * (1.f / 128.f);
    const float inv = rsqrtf(ss * (1.f / 128.f) - mu * mu + 1e-5f);
    for (int kc = 0; kc < 64; ++kc) {
      const int kcol = ch + kc;
      const float v = (Asub[r * 132 + kcol] - mu) * inv * lng[kcol] + lnb[kcol];
      xout[(size_t)(node0 + r) * HIDDIM + kcol] = fmaxf(v, 0.f);
    }
  }
}

// ----------------------------------------------------------- readout kernels
__global__ void colsum_kernel(const float* __restrict__ x,
                              float* __restrict__ g, int nrows) {
  __shared__ float part[2][HIDDIM];
  const int col = threadIdx.x & 127, grp = threadIdx.x >> 7;
  float s = 0.f;
  for (int r = blockIdx.x * 2 + grp; r < nrows; r += gridDim.x * 2)
    s += x[(size_t)r * HIDDIM + col];
  part[grp][col] = s;
  __syncthreads();
  if (grp == 0) atomicAdd(&g[col], part[0][col] + part[1][col]);
}

__global__ void final_mlp_kernel(const float* __restrict__ g,
                                 const float* __restrict__ W1,
                                 const float* __restrict__ b1,
                                 const float* __restrict__ W2,
                                 const float* __restrict__ b2,
                                 float* __restrict__ out, float invN) {
  __shared__ float gs[HIDDIM], hs[HIDDIM];
  const int t = threadIdx.x;
  gs[t] = g[t] * invN;
  __syncthreads();
  float acc = b1[t];
  for (int k = 0; k < HIDDIM; ++k) acc += gs[k] * W1[k * HIDDIM + t];
  hs[t] = fmaxf(acc, 0.f);
  __syncthreads();
  float o = b2[t];
  for (int k = 0; k < HIDDIM; ++k) o += hs[k] * W2[k * HIDDIM + t];
  out[t] = o;
}

// ------------------------------------------------------------------- launcher
extern "C" void kernel_launch(void* const* d_in, const int* in_sizes, int n_in,
                              void* d_out, int out_size, void* d_ws,
                              size_t ws_size, hipStream_t stream) {
  const float* node_feat = (const float*)d_in[0];
  const float* edge_attr = (const float*)d_in[1];
  const float* enc_W  = (const float*)d_in[2];
  const float* enc_b  = (const float*)d_in[3];
  const float* mlp_W1 = (const float*)d_in[4];
  const float* mlp_b1 = (const float*)d_in[5];
  const float* mlp_W2 = (const float*)d_in[6];
  const float* mlp_b2 = (const float*)d_in[7];
  const float* upd_W1 = (const float*)d_in[8];
  const float* upd_b1 = (const float*)d_in[9];
  const float* upd_W2 = (const float*)d_in[10];
  const float* upd_b2 = (const float*)d_in[11];
  const float* ln_g   = (const float*)d_in[12];
  const float* ln_b   = (const float*)d_in[13];
  const float* out_W1 = (const float*)d_in[14];
  const float* out_b1 = (const float*)d_in[15];
  const float* out_W2 = (const float*)d_in[16];
  const float* out_b2 = (const float*)d_in[17];
  const long long* edge_index = (const long long*)d_in[18];

  const int N = in_sizes[0] / 5;   // 50000
  const int E = in_sizes[1];       // 800000

  float* xa   = (float*)d_ws;
  float* xb   = xa + (size_t)N * HIDDIM;
  float* aggr = xb + (size_t)N * HIDDIM;
  float* invd = aggr + (size_t)N * HIDDIM;
  float* gbuf = invd + N;

  // degrees -> inv_deg (in place)
  zero_f32<<<(N + 255) / 256, 256, 0, stream>>>(invd, N);
  deg_acc_kernel<<<(E + 255) / 256, 256, 0, stream>>>(edge_index, invd, E);
  deg_inv_kernel<<<(N + 255) / 256, 256, 0, stream>>>(invd, N);
  // encoder
  encoder_kernel<<<(int)(((size_t)N * HIDDIM + 255) / 256), 256, 0, stream>>>(
      node_feat, enc_W, enc_b, xa, N);

  const size_t edge_shmem =
      (size_t)(128 * 260 + 128 * 132 + WAVES * 16 * 256 + WAVES * 16 * 132 +
               WAVES * 32 + WAVES * 16) * 4;  // ~300.8 KB (<= 320 KB WGP LDS)
  const size_t upd_shmem =
      (size_t)(128 * 260 + 128 * 132 + WAVES * 16 * 256 + WAVES * 16 * 132) * 4;

  float* xin = xa;
  float* xo  = xb;
  for (int l = 0; l < 3; ++l) {
    zero_f32<<<(int)(((size_t)N * HIDDIM + 255) / 256), 256, 0, stream>>>(
        aggr, (long long)N * HIDDIM);
    const int etiles = E / 16;
    int eblocks = (etiles + WAVES - 1) / WAVES;
    if (eblocks > 1024) eblocks = 1024;
    edge_msg_kernel<<<eblocks, WAVES * 32, edge_shmem, stream>>>(
        xin, edge_attr, edge_index,
        mlp_W1 + (size_t)l * 257 * HIDDIM, mlp_b1 + l * HIDDIM,
        mlp_W2 + (size_t)l * HIDDIM * HIDDIM, mlp_b2 + l * HIDDIM,
        invd, aggr, E);
    const int utiles = N / 16;
    int ublocks = (utiles + WAVES - 1) / WAVES;
    if (ublocks > 1024) ublocks = 1024;
    node_update_kernel<<<ublocks, WAVES * 32, upd_shmem, stream>>>(
        xin, aggr,
        upd_W1 + (size_t)l * 256 * HIDDIM, upd_b1 + l * HIDDIM,
        upd_W2 + (size_t)l * HIDDIM * HIDDIM, upd_b2 + l * HIDDIM,
        ln_g + l * HIDDIM, ln_b + l * HIDDIM, xo, N);
    float* t = xin; xin = xo; xo = t;
  }

  zero_f32<<<1, 128, 0, stream>>>(gbuf, 128);
  colsum_kernel<<<256, 256, 0, stream>>>(xin, gbuf, N);
  final_mlp_kernel<<<1, 128, 0, stream>>>(gbuf, out_W1, out_b1, out_W2, out_b2,
                                          (float*)d_out, 1.f / (float)N);
  (void)n_in; (void)out_size; (void)ws_size;
}